// attention_84834194030913
// MI455X (gfx1250) — compile-verified
//
#include <hip/hip_runtime.h>
#include <math.h>

#define NB   8
#define DIM  256
#define IMH  160
#define IMW  160
#define NBOX 512
#define SCALE 20
#define TOPK 5
#define MAXW (SCALE*SCALE*TOPK)   // 2000 tokens per image
#define CONF_T 0.2f
#define IOU_T  0.4f

typedef __attribute__((ext_vector_type(2))) float v2f;
typedef __attribute__((ext_vector_type(8))) float v8f;

// ---------------------------------------------------------------------------
// Kernel 1: per-image NMS + top-5 pick + patch-window clamp. One block/image.
// ---------------------------------------------------------------------------
__global__ __launch_bounds__(256) void nms_kernel(const float* __restrict__ rb,
                                                  float* __restrict__ out_mask,
                                                  float* __restrict__ out_boxes,
                                                  int* __restrict__ meta) {
    __shared__ float cf[NBOX];
    __shared__ float sx1[NBOX], sy1[NBOX], sx2[NBOX], sy2[NBOX];
    __shared__ int   keep[NBOX];
    __shared__ int   s_any;
    __shared__ int   sel[TOPK], bval[TOPK];
    __shared__ int   s_nk;

    const int b   = blockIdx.x;
    const int tid = threadIdx.x;
    const int bs  = blockDim.x;
    const float* boxes = rb + (size_t)b * NBOX * 5;

    if (tid == 0) s_any = 0;
    if (tid < TOPK) { sel[tid] = 0; bval[tid] = 0; }
    for (int i = tid; i < NBOX; i += bs) cf[i] = boxes[i * 5 + 4];
    __syncthreads();

    for (int i = tid; i < NBOX; i += bs)
        if (cf[i] > CONF_T) atomicOr(&s_any, 1);
    __syncthreads();
    const float thres = s_any ? CONF_T : 0.0f;

    // stable descending rank sort (matches jnp.argsort(-conf), stable)
    for (int i = tid; i < NBOX; i += bs) {
        const float ci = cf[i];
        int r = 0;
        for (int j = 0; j < NBOX; ++j) {
            const float cj = cf[j];
            r += (cj > ci) || (cj == ci && j < i);
        }
        sx1[r] = boxes[i * 5 + 0];
        sy1[r] = boxes[i * 5 + 1];
        sx2[r] = boxes[i * 5 + 2];
        sy2[r] = boxes[i * 5 + 3];
        keep[r] = (ci > thres) ? 1 : 0;
    }
    __syncthreads();

    // greedy forward NMS (equivalent to the reference fori_loop recurrence)
    for (int i = 0; i < NBOX - 1; ++i) {
        __syncthreads();
        if (!keep[i]) continue;
        const float ax1 = sx1[i], ay1 = sy1[i], ax2 = sx2[i], ay2 = sy2[i];
        const float aarea = (ax2 - ax1) * (ay2 - ay1);
        for (int j = i + 1 + tid; j < NBOX; j += bs) {
            if (!keep[j]) continue;
            const float ix1 = fmaxf(ax1, sx1[j]);
            const float iy1 = fmaxf(ay1, sy1[j]);
            const float ix2 = fminf(ax2, sx2[j]);
            const float iy2 = fminf(ay2, sy2[j]);
            const float inter = fmaxf(ix2 - ix1, 0.0f) * fmaxf(iy2 - iy1, 0.0f);
            const float barea = (sx2[j] - sx1[j]) * (sy2[j] - sy1[j]);
            const float iou = inter / (aarea + barea - inter + 1e-9f);
            if (iou > IOU_T) keep[j] = 0;
        }
    }
    __syncthreads();

    // size filter (applied after NMS in the reference)
    for (int i = tid; i < NBOX; i += bs)
        keep[i] = keep[i] && (sx2[i] - sx1[i] >= 1.0f) && (sy2[i] - sy1[i] >= 1.0f);
    __syncthreads();

    if (tid == 0) {
        int cnt = 0;
        for (int i = 0; i < NBOX && cnt < TOPK; ++i)
            if (keep[i]) { sel[cnt] = i; bval[cnt] = 1; ++cnt; }
        s_nk = cnt;
    }
    __syncthreads();

    if (tid < TOPK) {
        const int i = sel[tid];
        const int xi0 = (int)floorf(sx1[i] + 0.5f);
        const int yi0 = (int)floorf(sy1[i] + 0.5f);
        const int xi1 = (int)floorf(sx2[i] + 0.5f);
        const int yi1 = (int)floorf(sy2[i] + 0.5f);
        const float cx = (float)(xi0 + xi1) * 0.5f;
        const float cy = (float)(yi0 + yi1) * 0.5f;
        int x1 = (int)floorf(cx + 0.5f) - SCALE / 2;
        int y1 = (int)floorf(cy + 0.5f) - SCALE / 2;
        int x2 = x1 + SCALE, y2 = y1 + SCALE;
        if (x1 < 0) x2 = SCALE;
        x1 = max(x1, 0);
        if (y1 < 0) y2 = SCALE;
        y1 = max(y1, 0);
        if (x2 > IMW - 1) x1 = IMW - 1 - SCALE;
        x2 = min(x2, IMW - 1);
        if (y2 > IMH - 1) y1 = IMH - 1 - SCALE;
        y2 = min(y2, IMH - 1);
        const int s = b * TOPK + tid;
        meta[s * 3 + 0] = x1;
        meta[s * 3 + 1] = y1;
        meta[s * 3 + 2] = bval[tid];
        out_boxes[s * 4 + 0] = (float)x1;
        out_boxes[s * 4 + 1] = (float)y1;
        out_boxes[s * 4 + 2] = (float)x2;
        out_boxes[s * 4 + 3] = (float)y2;
    }
    __syncthreads();
    const int nk = s_nk;
    for (int j = tid; j < MAXW; j += bs)
        out_mask[(size_t)b * MAXW + j] = (j >= SCALE * SCALE * nk) ? 1.0f : 0.0f;
}

// ---------------------------------------------------------------------------
// Kernel 2: gather patches, tok = (feat + pos) * valid. One block per slot.
// ---------------------------------------------------------------------------
__global__ __launch_bounds__(256) void gather_kernel(const float* __restrict__ feat,
                                                     const float* __restrict__ pos,
                                                     const int* __restrict__ meta,
                                                     float* __restrict__ tok) {
    const int slot = blockIdx.x;          // b*TOPK + s
    const int b = slot / TOPK;
    const int x1 = meta[slot * 3 + 0];
    const int y1 = meta[slot * 3 + 1];
    const float v = (float)meta[slot * 3 + 2];
    const float* fb = feat + (size_t)b * DIM * IMH * IMW;
    const float* pb = pos + (size_t)b * DIM * IMH * IMW;
    float* tb = tok + (size_t)slot * SCALE * SCALE * DIM;
    for (int e = threadIdx.x; e < SCALE * SCALE * DIM; e += blockDim.x) {
        const int t = e >> 8;             // token in patch (y*20+x)
        const int d = e & (DIM - 1);
        const int ty = t / SCALE, tx = t % SCALE;
        const size_t off = ((size_t)d * IMH + (y1 + ty)) * IMW + (x1 + tx);
        tb[(size_t)t * DIM + d] = (fb[off] + pb[off]) * v;
    }
}

// ---------------------------------------------------------------------------
// Kernel 3/4: fp32 WMMA GEMM  C[M,N] = relu(A[M,K] @ W[K,N] + bias)
// one 16x16 output tile per wave, K stepped by 4 via V_WMMA_F32_16X16X4_F32
// ---------------------------------------------------------------------------
__global__ __launch_bounds__(256) void gemm_wmma(const float* __restrict__ A,
                                                 const float* __restrict__ Wm,
                                                 const float* __restrict__ bias,
                                                 float* __restrict__ C,
                                                 int M, int N, int K, int relu) {
    const int wave = threadIdx.x >> 5;
    const int lane = threadIdx.x & 31;
    const int half = lane >> 4;           // 0: lanes 0-15, 1: lanes 16-31
    const int l16  = lane & 15;
    const int tilesN = N >> 4;
    const int tile = blockIdx.x * 8 + wave;
    const int tm = tile / tilesN;
    const int tn = tile % tilesN;
    if (tm * 16 >= M) return;             // wave-uniform: EXEC stays all-ones

    const float* Arow = A + (size_t)(tm * 16 + l16) * K;
    const int col = tn * 16 + l16;
    v8f c = {};
    for (int k0 = 0; k0 < K; k0 += 4) {
        const int ka = k0 + half * 2;     // lanes 0-15: K=k0,k0+1; 16-31: K=k0+2,k0+3
        v2f a, bv;
        a.x = Arow[ka];
        a.y = Arow[ka + 1];
        bv.x = Wm[(size_t)ka * N + col];
        bv.y = Wm[(size_t)(ka + 1) * N + col];
        c = __builtin_amdgcn_wmma_f32_16x16x4_f32(false, a, false, bv,
                                                  (short)0, c, false, false);
    }
    const float bcol = bias[col];
    #pragma unroll
    for (int r = 0; r < 8; ++r) {
        const int row = tm * 16 + half * 8 + r;
        float val = c[r] + bcol;
        if (relu) val = fmaxf(val, 0.0f);
        C[(size_t)row * N + col] = val;
    }
}

// ---------------------------------------------------------------------------
// Kernel 5: seg = H2 @ w3 + b3, one wave32 per token
// ---------------------------------------------------------------------------
__global__ __launch_bounds__(256) void seg_kernel(const float* __restrict__ H2,
                                                  const float* __restrict__ w3,
                                                  const float* __restrict__ b3,
                                                  float* __restrict__ seg, int Mtot) {
    const int t = blockIdx.x * (blockDim.x >> 5) + (threadIdx.x >> 5);
    const int lane = threadIdx.x & 31;
    if (t >= Mtot) return;
    const float* h = H2 + (size_t)t * DIM;
    float s = 0.0f;
    for (int d = lane; d < DIM; d += 32) s += h[d] * w3[d];
    #pragma unroll
    for (int off = 16; off; off >>= 1) s += __shfl_xor(s, off, 32);
    if (lane == 0) seg[t] = s + b3[0];
}

// ---------------------------------------------------------------------------
extern "C" void kernel_launch(void* const* d_in, const int* in_sizes, int n_in,
                              void* d_out, int out_size, void* d_ws, size_t ws_size,
                              hipStream_t stream) {
    (void)in_sizes; (void)n_in; (void)out_size; (void)ws_size;
    const float* feat = (const float*)d_in[0];   // (B,D,H,W)
    const float* pos  = (const float*)d_in[1];   // (B,D,H,W)
    const float* rb   = (const float*)d_in[2];   // (B,NBOX,5)
    const float* w1   = (const float*)d_in[3];   // (D,D)
    const float* b1   = (const float*)d_in[4];   // (D,)
    const float* w2   = (const float*)d_in[5];   // (D,D)
    const float* b2   = (const float*)d_in[6];   // (D,)
    const float* w3   = (const float*)d_in[7];   // (D,1)
    const float* b3   = (const float*)d_in[8];   // (1,)

    float* out    = (float*)d_out;
    float* seg    = out;                         // B*MAXW
    float* omask  = out + NB * MAXW;             // B*MAXW
    float* oboxes = out + 2 * NB * MAXW;         // B*TOPK*4

    const int Mtot = NB * MAXW;                  // 16000 tokens
    char* ws = (char*)d_ws;
    int*   meta = (int*)ws;                                       // 120 ints
    float* tok  = (float*)(ws + 1024);                            // Mtot*DIM f32
    float* h1   = (float*)(ws + 1024 + (size_t)Mtot * DIM * 4);   // Mtot*DIM f32
    float* h2   = tok;                                            // reuse tok

    nms_kernel   <<<NB, 256, 0, stream>>>(rb, omask, oboxes, meta);
    gather_kernel<<<NB * TOPK, 256, 0, stream>>>(feat, pos, meta, tok);

    const int tiles = (Mtot / 16) * (DIM / 16);  // 16000 tiles, 8 waves/block
    gemm_wmma<<<tiles / 8, 256, 0, stream>>>(tok, w1, b1, h1, Mtot, DIM, DIM, 1);
    gemm_wmma<<<tiles / 8, 256, 0, stream>>>(h1,  w2, b2, h2, Mtot, DIM, DIM, 1);
    seg_kernel<<<(Mtot + 7) / 8, 256, 0, stream>>>(h2, w3, b3, seg, Mtot);
}